// DLI_loss_2_6614249636352
// MI455X (gfx1250) — compile-verified
//
#include <hip/hip_runtime.h>
#include <hip/hip_bf16.h>
#include <math.h>

// Problem constants (match reference)
#define BB 32
#define SS 4096
#define EE 256
#define HH 256
#define TT 64
#define JJ 62            // T - 2
#define MM (BB * JJ)     // 1984 LSTM rows; 1984 = 124 * 16 (exact 16-row tiles)

typedef __attribute__((ext_vector_type(2))) float v2f;
typedef __attribute__((ext_vector_type(8))) float v8f;

__device__ __forceinline__ float sigm_fast(float x) {
    return 1.0f / (1.0f + __expf(-x));
}
__device__ __forceinline__ float tanh_fast(float x) {
    x = fminf(fmaxf(x, -15.0f), 15.0f);
    float e = __expf(2.0f * x);
    return (e - 1.0f) / (e + 1.0f);
}

// ---------------------------------------------------------------------------
// K1: segment means (bandwidth-dominant: 128 MB read) + t = states . We
// grid = B*T blocks of 256 threads. Per block: one 64-row x 256-col segment.
//   thread = rowgroup(4) x colthread(64); each thread streams 16 rows of a
//   float4 column slice -> 4 independent b128 load chains per column for MLP.
//   global_prefetch_b8 issued ~8 rows ahead of the stream.
// ---------------------------------------------------------------------------
__global__ void seg_mean_kernel(const float* __restrict__ enc,
                                const int*   __restrict__ ends,
                                const float* __restrict__ fc_w,
                                float* __restrict__ states,
                                float* __restrict__ tproj)
{
    __shared__ float4 red4[4][64];
    __shared__ float  redp[64];

    int seg = blockIdx.x;           // b*T + tau
    int b   = seg >> 6;             // T == 64
    int tau = seg & 63;
    int e   = ends[b * TT + tau];
    int s0  = (tau == 0) ? 0 : (ends[b * TT + tau - 1] + 1);
    int len = e - s0 + 1;

    int rg = threadIdx.x >> 6;      // row group 0..3
    int ct = threadIdx.x & 63;      // column thread 0..63
    int e4 = ct * 4;                // 64 threads * 4 = 256 columns

    // row range for this row group (len==64 in practice; stay generic)
    int r0 = (len * rg) / 4;
    int r1 = (len * (rg + 1)) / 4;

    const float* base = enc + ((size_t)b * SS + (size_t)(s0 + r0)) * EE + e4;
    float4 acc = make_float4(0.f, 0.f, 0.f, 0.f);
    for (int s = r0; s < r1; ++s) {
        __builtin_prefetch(base + 8 * EE, 0, 1);   // global_prefetch_b8, 8 rows ahead
        float4 v = *(const float4*)base;           // global_load_b128, coalesced
        acc.x += v.x; acc.y += v.y; acc.z += v.z; acc.w += v.w;
        base += EE;
    }
    red4[rg][ct] = acc;
    __syncthreads();

    if (rg == 0) {
        float4 a0 = red4[0][ct], a1 = red4[1][ct], a2 = red4[2][ct], a3 = red4[3][ct];
        float inv = 1.0f / (float)len;
        float4 m;
        m.x = (a0.x + a1.x + a2.x + a3.x) * inv;
        m.y = (a0.y + a1.y + a2.y + a3.y) * inv;
        m.z = (a0.z + a1.z + a2.z + a3.z) * inv;
        m.w = (a0.w + a1.w + a2.w + a3.w) * inv;
        *(float4*)(states + (size_t)seg * EE + e4) = m;

        // projection onto We = fc_w[H : H+E]
        redp[ct] = m.x * fc_w[HH + e4]     + m.y * fc_w[HH + e4 + 1]
                 + m.z * fc_w[HH + e4 + 2] + m.w * fc_w[HH + e4 + 3];
    }
    __syncthreads();
    for (int off = 32; off > 0; off >>= 1) {
        if (threadIdx.x < off) redp[threadIdx.x] += redp[threadIdx.x + off];
        __syncthreads();
    }
    if (threadIdx.x == 0) tproj[seg] = redp[0];
}

// ---------------------------------------------------------------------------
// LSTM GEMM tiles via V_WMMA_F32_16X16X4_F32 (fp32 WMMA, wave32).
// One wave computes one 16x16 output tile for ALL FOUR gates (4 accumulators),
// so i/f/g/o live in the same lanes and activations fuse in registers.
//   A 16x4 frag : lane<16 -> (M=lane, K=kk+0/1), lane>=16 -> (M=lane-16, K=kk+2/3)
//   B 4x16 frag : same pattern with N=lane&15 (B[k][n] = W[n][k], W row-major)
//   C/D 16x16   : vgpr r -> M = r + 8*(lane>>4), N = lane&15
// ---------------------------------------------------------------------------
__global__ void lstm_cell1_kernel(const float* __restrict__ states,
                                  const float* __restrict__ W_ih,
                                  const float* __restrict__ b_ih,
                                  const float* __restrict__ b_hh,
                                  float* __restrict__ h1,
                                  float* __restrict__ c1)
{
    int wave = blockIdx.x * (blockDim.x >> 5) + (threadIdx.x >> 5);
    int lane = threadIdx.x & 31;
    const int NT = HH / 16;                 // 16 n-tiles per gate block
    int mt = wave / NT, nt = wave - mt * NT;
    int m0 = mt * 16, n0 = nt * 16;
    int half = lane >> 4;
    int l15  = lane & 15;
    int koff = half * 2;

    // A (x1): row m0+l15 maps to state (b, j)
    int mrow = m0 + l15;
    int brow = mrow / JJ;
    int jrow = mrow - brow * JJ;
    const float* xrow = states + (size_t)(brow * TT + jrow) * EE;

    // B: W_ih rows (gate*H + n0 + l15)
    const float* wrow0 = W_ih + (size_t)(0 * HH + n0 + l15) * EE;
    const float* wrow1 = W_ih + (size_t)(1 * HH + n0 + l15) * EE;
    const float* wrow2 = W_ih + (size_t)(2 * HH + n0 + l15) * EE;
    const float* wrow3 = W_ih + (size_t)(3 * HH + n0 + l15) * EE;

    v8f acc0 = {0,0,0,0,0,0,0,0};
    v8f acc1 = {0,0,0,0,0,0,0,0};
    v8f acc2 = {0,0,0,0,0,0,0,0};
    v8f acc3 = {0,0,0,0,0,0,0,0};

    for (int kk = 0; kk < EE; kk += 4) {
        v2f a   = *(const v2f*)(xrow  + kk + koff);
        v2f bf0 = *(const v2f*)(wrow0 + kk + koff);
        v2f bf1 = *(const v2f*)(wrow1 + kk + koff);
        v2f bf2 = *(const v2f*)(wrow2 + kk + koff);
        v2f bf3 = *(const v2f*)(wrow3 + kk + koff);
        acc0 = __builtin_amdgcn_wmma_f32_16x16x4_f32(false, a, false, bf0, (short)0, acc0, false, false);
        acc1 = __builtin_amdgcn_wmma_f32_16x16x4_f32(false, a, false, bf1, (short)0, acc1, false, false);
        acc2 = __builtin_amdgcn_wmma_f32_16x16x4_f32(false, a, false, bf2, (short)0, acc2, false, false);
        acc3 = __builtin_amdgcn_wmma_f32_16x16x4_f32(false, a, false, bf3, (short)0, acc3, false, false);
    }

    int ncol = n0 + l15;
    float bi  = b_ih[0 * HH + ncol] + b_hh[0 * HH + ncol];
    float bff = b_ih[1 * HH + ncol] + b_hh[1 * HH + ncol];
    float bg  = b_ih[2 * HH + ncol] + b_hh[2 * HH + ncol];
    float bo  = b_ih[3 * HH + ncol] + b_hh[3 * HH + ncol];

#pragma unroll
    for (int r = 0; r < 8; ++r) {
        int m = m0 + r + half * 8;
        float gi = acc0[r] + bi;
        float gf = acc1[r] + bff;   // c_prev == 0 -> sigmoid(f)*0 contributes 0 (DCE'd)
        float gg = acc2[r] + bg;
        float go = acc3[r] + bo;
        (void)gf;
        float c = sigm_fast(gi) * tanh_fast(gg);
        float h = sigm_fast(go) * tanh_fast(c);
        c1[(size_t)m * HH + ncol] = c;
        h1[(size_t)m * HH + ncol] = h;
    }
}

__global__ void lstm_cell2_kernel(const float* __restrict__ states,
                                  const float* __restrict__ h1,
                                  const float* __restrict__ c1,
                                  const float* __restrict__ W_ih,
                                  const float* __restrict__ W_hh,
                                  const float* __restrict__ b_ih,
                                  const float* __restrict__ b_hh,
                                  float* __restrict__ h2)
{
    int wave = blockIdx.x * (blockDim.x >> 5) + (threadIdx.x >> 5);
    int lane = threadIdx.x & 31;
    const int NT = HH / 16;
    int mt = wave / NT, nt = wave - mt * NT;
    int m0 = mt * 16, n0 = nt * 16;
    int half = lane >> 4;
    int l15  = lane & 15;
    int koff = half * 2;

    int mrow = m0 + l15;
    int brow = mrow / JJ;
    int jrow = mrow - brow * JJ;
    const float* xrow = states + (size_t)(brow * TT + jrow + 1) * EE;  // x2 = states[:,1:J+1]
    const float* hrow = h1 + (size_t)mrow * HH;

    const float* wi0 = W_ih + (size_t)(0 * HH + n0 + l15) * EE;
    const float* wi1 = W_ih + (size_t)(1 * HH + n0 + l15) * EE;
    const float* wi2 = W_ih + (size_t)(2 * HH + n0 + l15) * EE;
    const float* wi3 = W_ih + (size_t)(3 * HH + n0 + l15) * EE;
    const float* wh0 = W_hh + (size_t)(0 * HH + n0 + l15) * HH;
    const float* wh1 = W_hh + (size_t)(1 * HH + n0 + l15) * HH;
    const float* wh2 = W_hh + (size_t)(2 * HH + n0 + l15) * HH;
    const float* wh3 = W_hh + (size_t)(3 * HH + n0 + l15) * HH;

    v8f acc0 = {0,0,0,0,0,0,0,0};
    v8f acc1 = {0,0,0,0,0,0,0,0};
    v8f acc2 = {0,0,0,0,0,0,0,0};
    v8f acc3 = {0,0,0,0,0,0,0,0};

    for (int kk = 0; kk < EE; kk += 4) {       // x2 @ W_ih^T
        v2f a  = *(const v2f*)(xrow + kk + koff);
        v2f b0 = *(const v2f*)(wi0 + kk + koff);
        v2f b1 = *(const v2f*)(wi1 + kk + koff);
        v2f b2 = *(const v2f*)(wi2 + kk + koff);
        v2f b3 = *(const v2f*)(wi3 + kk + koff);
        acc0 = __builtin_amdgcn_wmma_f32_16x16x4_f32(false, a, false, b0, (short)0, acc0, false, false);
        acc1 = __builtin_amdgcn_wmma_f32_16x16x4_f32(false, a, false, b1, (short)0, acc1, false, false);
        acc2 = __builtin_amdgcn_wmma_f32_16x16x4_f32(false, a, false, b2, (short)0, acc2, false, false);
        acc3 = __builtin_amdgcn_wmma_f32_16x16x4_f32(false, a, false, b3, (short)0, acc3, false, false);
    }
    for (int kk = 0; kk < HH; kk += 4) {       // h1 @ W_hh^T
        v2f a  = *(const v2f*)(hrow + kk + koff);
        v2f b0 = *(const v2f*)(wh0 + kk + koff);
        v2f b1 = *(const v2f*)(wh1 + kk + koff);
        v2f b2 = *(const v2f*)(wh2 + kk + koff);
        v2f b3 = *(const v2f*)(wh3 + kk + koff);
        acc0 = __builtin_amdgcn_wmma_f32_16x16x4_f32(false, a, false, b0, (short)0, acc0, false, false);
        acc1 = __builtin_amdgcn_wmma_f32_16x16x4_f32(false, a, false, b1, (short)0, acc1, false, false);
        acc2 = __builtin_amdgcn_wmma_f32_16x16x4_f32(false, a, false, b2, (short)0, acc2, false, false);
        acc3 = __builtin_amdgcn_wmma_f32_16x16x4_f32(false, a, false, b3, (short)0, acc3, false, false);
    }

    int ncol = n0 + l15;
    float bi  = b_ih[0 * HH + ncol] + b_hh[0 * HH + ncol];
    float bff = b_ih[1 * HH + ncol] + b_hh[1 * HH + ncol];
    float bg  = b_ih[2 * HH + ncol] + b_hh[2 * HH + ncol];
    float bo  = b_ih[3 * HH + ncol] + b_hh[3 * HH + ncol];

#pragma unroll
    for (int r = 0; r < 8; ++r) {
        int m = m0 + r + half * 8;
        float gi = acc0[r] + bi;
        float gf = acc1[r] + bff;
        float gg = acc2[r] + bg;
        float go = acc3[r] + bo;
        float cp = c1[(size_t)m * HH + ncol];
        float c  = sigm_fast(gf) * cp + sigm_fast(gi) * tanh_fast(gg);
        float h  = sigm_fast(go) * tanh_fast(c);
        h2[(size_t)m * HH + ncol] = h;
    }
}

// ---------------------------------------------------------------------------
// K4: one wave32 per (b,j): a = h2 . Wh, masked logsumexp over k, loss[b,j]
// ---------------------------------------------------------------------------
__global__ void loss_kernel(const float* __restrict__ h2,
                            const float* __restrict__ tproj,
                            const float* __restrict__ fc_w,
                            const float* __restrict__ fc_b,
                            float* __restrict__ loss)
{
    int wave = blockIdx.x * (blockDim.x >> 5) + (threadIdx.x >> 5);
    int lane = threadIdx.x & 31;
    if (wave >= MM) return;
    int b = wave / JJ, j = wave - b * JJ;

    const float* hrow = h2 + (size_t)wave * HH;
    float p = 0.f;
    for (int e = lane; e < HH; e += 32) p += hrow[e] * fc_w[e];   // Wh part
    for (int off = 16; off > 0; off >>= 1) p += __shfl_xor(p, off, 32);
    float afull = p + fc_b[0];

    const float* tb = tproj + b * TT;
    int kmax = JJ - 1 - j;                 // k valid iff j+2+k < T
    float v0 = (lane      <= kmax) ? (afull + tb[j + 2 + lane])      : -1e30f;
    float v1 = (lane + 32 <= kmax) ? (afull + tb[j + 2 + lane + 32]) : -1e30f;
    float mx = fmaxf(v0, v1);
    for (int off = 16; off > 0; off >>= 1) mx = fmaxf(mx, __shfl_xor(mx, off, 32));
    float s = 0.f;
    if (lane      <= kmax) s += __expf(v0 - mx);
    if (lane + 32 <= kmax) s += __expf(v1 - mx);
    for (int off = 16; off > 0; off >>= 1) s += __shfl_xor(s, off, 32);
    if (lane == 0) {
        float lse = mx + __logf(s);
        loss[wave] = lse - (afull + tb[j + 2]);   // logits[...,0]
    }
}

__global__ void mean_kernel(const float* __restrict__ loss, float* __restrict__ out)
{
    __shared__ float red[256];
    float s = 0.f;
    for (int i = threadIdx.x; i < MM; i += 256) s += loss[i];
    red[threadIdx.x] = s;
    __syncthreads();
    for (int off = 128; off > 0; off >>= 1) {
        if (threadIdx.x < off) red[threadIdx.x] += red[threadIdx.x + off];
        __syncthreads();
    }
    if (threadIdx.x == 0) out[0] = red[0] / (float)MM;
}

// ---------------------------------------------------------------------------
extern "C" void kernel_launch(void* const* d_in, const int* in_sizes, int n_in,
                              void* d_out, int out_size, void* d_ws, size_t ws_size,
                              hipStream_t stream)
{
    (void)in_sizes; (void)n_in; (void)out_size; (void)ws_size;
    const float* enc  = (const float*)d_in[0];
    const int*   ends = (const int*)  d_in[1];
    const float* W_ih = (const float*)d_in[2];
    const float* W_hh = (const float*)d_in[3];
    const float* b_ih = (const float*)d_in[4];
    const float* b_hh = (const float*)d_in[5];
    const float* fc_w = (const float*)d_in[6];
    const float* fc_b = (const float*)d_in[7];
    float* out = (float*)d_out;

    // workspace layout (floats)
    float* ws     = (float*)d_ws;
    float* states = ws;                            // B*T*E   = 524288
    float* tproj  = states + (size_t)BB * TT * EE; //            2048
    float* h1     = tproj  + (size_t)BB * TT;      // M*H     = 507904
    float* c1     = h1     + (size_t)MM * HH;      //           507904
    float* h2     = c1     + (size_t)MM * HH;      //           507904
    float* lossv  = h2     + (size_t)MM * HH;      //             1984

    // K1: segment means + t projection (128 MB read -> HBM-bound, ~5.5us floor)
    seg_mean_kernel<<<BB * TT, 256, 0, stream>>>(enc, ends, fc_w, states, tproj);

    // K2/K3: LSTM cells; (M/16)*(H/16) = 124*16 = 1984 wave-tiles, 8 waves/block
    const int lstm_blocks = (MM / 16) * (HH / 16) / 8;   // 248
    lstm_cell1_kernel<<<lstm_blocks, 256, 0, stream>>>(states, W_ih, b_ih, b_hh, h1, c1);
    lstm_cell2_kernel<<<lstm_blocks, 256, 0, stream>>>(states, h1, c1, W_ih, W_hh,
                                                       b_ih, b_hh, h2);

    // K4: one wave per (b,j); 1984 waves -> 248 blocks of 256 threads
    loss_kernel<<<(MM + 7) / 8, 256, 0, stream>>>(h2, tproj, fc_w, fc_b, lossv);

    // K5: mean over 1984 losses -> scalar
    mean_kernel<<<1, 256, 0, stream>>>(lossv, out);
}